// GraphGPSLayer_63874753626522
// MI455X (gfx1250) — compile-verified
//
#include <hip/hip_runtime.h>
#include <hip/hip_bf16.h>

// ---------------------------------------------------------------------------
// GraphGPS layer for MI455X (gfx1250, wave32).
// Dense GEMMs use V_WMMA_F32_16X16X4_F32 (fp32 matrix pipe, exact fp32),
// with the B (weight) panel staged through LDS and A software-pipelined.
// Edge gather/scatter phases (the bandwidth bound) use float atomics.
// ---------------------------------------------------------------------------

typedef __attribute__((ext_vector_type(2))) float v2f;
typedef __attribute__((ext_vector_type(8))) float v8f;

#define H 128
#define NHEAD 8
#define DHEAD 16

__device__ __forceinline__ float atomicAddF(float* p, float v) {
  return __hip_atomic_fetch_add(p, v, __ATOMIC_RELAXED, __HIP_MEMORY_SCOPE_AGENT);
}

__device__ __forceinline__ void atomicMaxF(float* addr, float v) {
  // Standard ordered-int trick: works for any mix of signs.
  if (v >= 0.0f) atomicMax((int*)addr, __float_as_int(v));
  else           atomicMin((unsigned int*)addr, __float_as_uint(v));
}

// ---------------------------------------------------------------------------
// Generic GEMM: C[m,n] = epilogue( alpha * (A[m,:] . W[n,:] + bias[n]) + res )
// A: [nrows, K] row-major, W: [OUT, K] row-major (so C = A @ W^T).
// Block = 256 threads (8 waves); block covers 128 rows x 128 cols.
// W panel (128 x K) is staged in LDS (stride K+4 to spread banks);
// each wave computes a 16-row strip via v_wmma_f32_16x16x4_f32.
// Operand layout per ISA 7.12.2:
//   A 16x4 : lanes 0-15 -> M=lane, V0=K0 V1=K1 ; lanes 16-31 -> V0=K2 V1=K3
//   B 4x16 : lanes 0-15 -> N=lane, K0/K1      ; lanes 16-31 -> N=lane-16, K2/K3
//   C/D    : VGPR r -> M = r (+8 for upper half), N = lane&15
// ---------------------------------------------------------------------------
__global__ __launch_bounds__(256) void gemm_wmma_f32(
    const float* __restrict__ A, const float* __restrict__ Wm,
    const float* __restrict__ bias, const float* __restrict__ res,
    float* __restrict__ C, int nrows, int K, int OUT, float alpha, int relu)
{
  extern __shared__ float sW[];                   // [128][K+4]
  const int tid     = threadIdx.x;
  const int lane    = tid & 31;
  const int wave    = tid >> 5;
  const int colbase = blockIdx.y * 128;
  const int ldW     = K + 4;
  const int kvec    = K >> 2;                     // float4s per W row
  const int kvShift = (K == 128) ? 5 : 6;         // log2(kvec)

  // ---- cooperative stage of W[colbase .. colbase+127][0..K) into LDS ----
  for (int i = tid; i < (kvec << 7); i += 256) {
    const int r  = i >> kvShift;
    const int kv = (i & (kvec - 1)) << 2;
    float4 w4 = *(const float4*)(Wm + (size_t)(colbase + r) * K + kv);
    *(float4*)(&sW[r * ldW + kv]) = w4;           // ldW % 4 == 0 -> 16B aligned
  }
  __syncthreads();

  const int strip = blockIdx.x * 8 + wave;        // 16-row strip index
  const int row0  = strip * 16;
  if (row0 >= nrows) return;                      // after barrier; wave-uniform

  const int l    = lane & 15;
  const int half = lane >> 4;

  v8f acc[8];
  const v8f zero = {0.f,0.f,0.f,0.f,0.f,0.f,0.f,0.f};
#pragma unroll
  for (int i = 0; i < 8; ++i) acc[i] = zero;

  const float* arow = A + (size_t)(row0 + l) * K;

  // software pipeline: a for kb=0 preloaded, a(kb+4) fetched before WMMA burst
  v2f a;
  a.x = arow[2 * half + 0];
  a.y = arow[2 * half + 1];

  for (int kb = 0; kb < K; kb += 4) {
    const int kc = kb + 2 * half;

    // batched LDS reads of the 8 B operands for this k-step
    v2f b[8];
#pragma unroll
    for (int ct = 0; ct < 8; ++ct) {
      const float* wp = &sW[(ct * 16 + l) * ldW + kc];
      b[ct].x = wp[0];
      b[ct].y = wp[1];
    }

    // prefetch next A operand from global while WMMAs run
    v2f an = a;
    if (kb + 4 < K) {
      an.x = arow[kc + 4];
      an.y = arow[kc + 5];
    }

#pragma unroll
    for (int ct = 0; ct < 8; ++ct) {
      acc[ct] = __builtin_amdgcn_wmma_f32_16x16x4_f32(
          /*neg_a=*/false, a, /*neg_b=*/false, b[ct],
          /*c_mod=*/(short)0, acc[ct], /*reuse_a=*/false, /*reuse_b=*/false);
    }
    a = an;
  }

  // ---- epilogue ----
#pragma unroll
  for (int ct = 0; ct < 8; ++ct) {
    const int n  = colbase + ct * 16 + l;
    const float bn = bias ? bias[n] : 0.0f;
#pragma unroll
    for (int r = 0; r < 8; ++r) {
      const int m = row0 + r + 8 * half;
      if (m < nrows) {
        float val = (acc[ct][r] + bn) * alpha;
        if (res)  val += res[(size_t)m * OUT + n];
        if (relu) val = fmaxf(val, 0.0f);
        C[(size_t)m * OUT + n] = val;
      }
    }
  }
}

// --------------------------- edge-phase kernels ----------------------------

// GINE message + scatter-add: agg[dst] += relu(h[src] + eattr)
__global__ __launch_bounds__(256) void edge_gine(
    const float* __restrict__ h, const float* __restrict__ ea,
    const int* __restrict__ src, const int* __restrict__ dst,
    float* __restrict__ agg, int E)
{
  int idx = blockIdx.x * 256 + threadIdx.x;       // E * 32 threads, 4 floats each
  int e = idx >> 5;
  if (e >= E) return;
  int c4 = (idx & 31) << 2;
  int s = src[e], d = dst[e];
  float4 hv = *(const float4*)(h + (size_t)s * H + c4);
  float4 ev = *(const float4*)(ea + (size_t)e * H + c4);
  float* ap = agg + (size_t)d * H + c4;
  atomicAddF(ap + 0, fmaxf(hv.x + ev.x, 0.f));
  atomicAddF(ap + 1, fmaxf(hv.y + ev.y, 0.f));
  atomicAddF(ap + 2, fmaxf(hv.z + ev.z, 0.f));
  atomicAddF(ap + 3, fmaxf(hv.w + ev.w, 0.f));
}

// per-edge, per-head attention score + running row max over src
__global__ __launch_bounds__(256) void edge_score(
    const float* __restrict__ q, const float* __restrict__ k,
    const int* __restrict__ src, const int* __restrict__ dst,
    float* __restrict__ sbuf, float* __restrict__ mbuf, int E)
{
  int idx = blockIdx.x * 256 + threadIdx.x;       // E * NHEAD threads
  int e = idx >> 3;
  if (e >= E) return;
  int hd = idx & 7;
  const float* qp = q + (size_t)src[e] * H + hd * DHEAD;
  const float* kp = k + (size_t)dst[e] * H + hd * DHEAD;
  float acc = 0.f;
#pragma unroll
  for (int j = 0; j < DHEAD; j += 4) {
    float4 a = *(const float4*)(qp + j);
    float4 b = *(const float4*)(kp + j);
    acc += a.x * b.x + a.y * b.y + a.z * b.z + a.w * b.w;
  }
  sbuf[idx] = acc;
  atomicMaxF(&mbuf[(size_t)src[e] * NHEAD + hd], acc);
}

// p = exp(s - m[src]);  denom[src] += p
__global__ __launch_bounds__(256) void edge_exp(
    float* __restrict__ sbuf, const float* __restrict__ mbuf,
    const int* __restrict__ src, float* __restrict__ denom, int E)
{
  int idx = blockIdx.x * 256 + threadIdx.x;
  int e = idx >> 3;
  if (e >= E) return;
  int hd = idx & 7;
  float p = expf(sbuf[idx] - mbuf[(size_t)src[e] * NHEAD + hd]);
  sbuf[idx] = p;
  atomicAddF(&denom[(size_t)src[e] * NHEAD + hd], p);
}

// out[src] += (p / denom[src]) * v[dst]
__global__ __launch_bounds__(256) void edge_out(
    const float* __restrict__ sbuf, const float* __restrict__ denom,
    const float* __restrict__ v, const int* __restrict__ src,
    const int* __restrict__ dst, float* __restrict__ outb, int E)
{
  int idx = blockIdx.x * 256 + threadIdx.x;       // E * 32 threads
  int e = idx >> 5;
  if (e >= E) return;
  int rem = idx & 31;
  int hd  = rem >> 2;
  int c4  = (rem & 3) << 2;
  int s = src[e], d = dst[e];
  float w = sbuf[(size_t)e * NHEAD + hd] / denom[(size_t)s * NHEAD + hd];
  float4 vv = *(const float4*)(v + (size_t)d * H + hd * DHEAD + c4);
  float* op = outb + (size_t)s * H + hd * DHEAD + c4;
  atomicAddF(op + 0, w * vv.x);
  atomicAddF(op + 1, w * vv.y);
  atomicAddF(op + 2, w * vv.z);
  atomicAddF(op + 3, w * vv.w);
}

// --------------------------- BatchNorm helpers -----------------------------

// Per-column sum/sumsq over rows; optionally x = a + b stored to xout.
__global__ __launch_bounds__(256) void bn_stats(
    const float* __restrict__ a, const float* __restrict__ b,
    float* __restrict__ xout, float* __restrict__ gsum,
    float* __restrict__ gsq, int nrows)
{
  int c    = threadIdx.x & 127;
  int rsub = threadIdx.x >> 7;                    // 0..1
  int r0   = blockIdx.x * 64;
  int rend = r0 + 64; if (rend > nrows) rend = nrows;
  float s = 0.f, s2 = 0.f;
  for (int r = r0 + rsub; r < rend; r += 2) {
    float x = a[(size_t)r * H + c];
    if (b) {
      x += b[(size_t)r * H + c];
      if (xout) xout[(size_t)r * H + c] = x;
    }
    s += x; s2 += x * x;
  }
  atomicAddF(&gsum[c], s);
  atomicAddF(&gsq[c], s2);
}

// scale/shift so that xn = x*sc + sh == (x-mu)*rsqrt(var+eps)*g + b
__global__ void bn_finalize(
    const float* __restrict__ sum, const float* __restrict__ sq,
    const float* __restrict__ g, const float* __restrict__ b,
    float* __restrict__ sc, float* __restrict__ sh, int n)
{
  int c = threadIdx.x;
  float inv_n = 1.0f / (float)n;
  float mu  = sum[c] * inv_n;
  float var = sq[c] * inv_n - mu * mu;
  float inv = rsqrtf(var + 1e-5f);
  float s   = inv * g[c];
  sc[c] = s;
  sh[c] = b[c] - mu * s;
}

// --------------------------- elementwise kernels ---------------------------

__global__ __launch_bounds__(256) void ew_add_inplace(
    float* __restrict__ a, const float* __restrict__ b, int n)
{
  int i = blockIdx.x * 256 + threadIdx.x;
  if (i < n) a[i] += b[i];
}

__global__ __launch_bounds__(256) void ew_fill(float* __restrict__ p, float v, int n)
{
  int i = blockIdx.x * 256 + threadIdx.x;
  if (i < n) p[i] = v;
}

// hc = bn0(x1) + bn1(x2)
__global__ __launch_bounds__(256) void ew_combine(
    const float* __restrict__ x1, const float* __restrict__ x2,
    const float* __restrict__ sc0, const float* __restrict__ sh0,
    const float* __restrict__ sc1, const float* __restrict__ sh1,
    float* __restrict__ hc, int n)
{
  int i = blockIdx.x * 256 + threadIdx.x;
  if (i >= n) return;
  int c = i & 127;
  hc[i] = (x1[i] * sc0[c] + sh0[c]) + (x2[i] * sc1[c] + sh1[c]);
}

// out = bn(x)
__global__ __launch_bounds__(256) void ew_apply_bn(
    const float* __restrict__ x, const float* __restrict__ sc,
    const float* __restrict__ sh, float* __restrict__ out, int n)
{
  int i = blockIdx.x * 256 + threadIdx.x;
  if (i >= n) return;
  int c = i & 127;
  out[i] = x[i] * sc[c] + sh[c];
}

// ---------------------------------------------------------------------------

extern "C" void kernel_launch(void* const* d_in, const int* in_sizes, int n_in,
                              void* d_out, int out_size, void* d_ws, size_t ws_size,
                              hipStream_t stream) {
  const float* h      = (const float*)d_in[0];
  const float* eattr  = (const float*)d_in[1];
  const int*   src    = (const int*)  d_in[2];
  const int*   dst    = (const int*)  d_in[3];
  const float* gin_w1 = (const float*)d_in[4];
  const float* gin_b1 = (const float*)d_in[5];
  const float* gin_w2 = (const float*)d_in[6];
  const float* gin_b2 = (const float*)d_in[7];
  const float* wq     = (const float*)d_in[8];
  const float* bq     = (const float*)d_in[9];
  const float* wk     = (const float*)d_in[10];
  const float* bk     = (const float*)d_in[11];
  const float* wv     = (const float*)d_in[12];
  const float* bv     = (const float*)d_in[13];
  const float* nl_g   = (const float*)d_in[14];
  const float* nl_b   = (const float*)d_in[15];
  const float* na_g   = (const float*)d_in[16];
  const float* na_b   = (const float*)d_in[17];
  const float* no_g   = (const float*)d_in[18];
  const float* no_b   = (const float*)d_in[19];
  const float* ffn1_w = (const float*)d_in[20];
  const float* ffn1_b = (const float*)d_in[21];
  const float* ffn2_w = (const float*)d_in[22];
  const float* ffn2_b = (const float*)d_in[23];

  const int N = in_sizes[0] / H;
  const int E = in_sizes[2];
  const size_t NHf = (size_t)N * H;

  // -------- workspace layout (floats) --------
  float* Wsp  = (float*)d_ws;
  float* agg  = Wsp;                 // [N,H]  GINE agg, then z = h+agg (in place)
  float* tmp1 = Wsp + 1 * NHf;       // [N,H]  relu MLP hidden; later x3 (pre-BN out)
  float* x1   = Wsp + 2 * NHf;       // [N,H]  local branch pre-BN
  float* qb   = Wsp + 3 * NHf;       // [N,H]
  float* kb   = Wsp + 4 * NHf;       // [N,H]
  float* vb   = Wsp + 5 * NHf;       // [N,H]
  float* x2   = Wsp + 6 * NHf;       // [N,H]  attention accum, then +h (pre-BN)
  float* hcb  = Wsp + 7 * NHf;       // [N,H]
  float* f1b  = Wsp + 8 * NHf;       // [N,2H]
  float* sbuf = Wsp + 10 * NHf;      // [E,NHEAD] scores -> p
  float* mbuf = sbuf + (size_t)E * NHEAD;   // [N,NHEAD]
  float* dbuf = mbuf + (size_t)N * NHEAD;   // [N,NHEAD]
  float* st   = dbuf + (size_t)N * NHEAD;   // 3 BN sets x {sum,sq,sc,sh} x 128
  float* sum0 = st + 0*512, *sq0 = sum0+128, *sc0 = sum0+256, *sh0 = sum0+384;
  float* sum1 = st + 1*512, *sq1 = sum1+128, *sc1 = sum1+256, *sh1 = sum1+384;
  float* sum2 = st + 2*512, *sq2 = sum2+128, *sc2 = sum2+256, *sh2 = sum2+384;

  float* outp = (float*)d_out;

  // LDS panel sizes for the GEMM (K+4 padded stride), opt-in above 64KB.
  const int shK128 = 128 * (H + 4) * (int)sizeof(float);          //  67.6 KB
  const int shK256 = 128 * (2 * H + 4) * (int)sizeof(float);      // 133.1 KB
  hipFuncSetAttribute((const void*)gemm_wmma_f32,
                      hipFuncAttributeMaxDynamicSharedMemorySize, shK256);

  // -------- init --------
  hipMemsetAsync(agg,  0, NHf * sizeof(float), stream);
  hipMemsetAsync(x2,   0, NHf * sizeof(float), stream);
  hipMemsetAsync(dbuf, 0, (size_t)N * NHEAD * sizeof(float), stream);
  hipMemsetAsync(st,   0, 3 * 512 * sizeof(float), stream);
  ew_fill<<<(N * NHEAD + 255) / 256, 256, 0, stream>>>(mbuf, -__builtin_inff(), N * NHEAD);

  const int strips = (N + 15) / 16;
  const int ggx    = (strips + 7) / 8;
  const int ewB    = (int)((NHf + 255) / 256);
  const int bnB    = (N + 63) / 64;
  const int eh8    = (E * NHEAD + 255) / 256;
  const int eh32   = (E * 32 + 255) / 256;

  // -------- GINE local branch --------
  edge_gine<<<eh32, 256, 0, stream>>>(h, eattr, src, dst, agg, E);
  ew_add_inplace<<<ewB, 256, 0, stream>>>(agg, h, (int)NHf);            // z = h + agg
  gemm_wmma_f32<<<dim3(ggx, 1), 256, shK128, stream>>>(agg, gin_w1, gin_b1, nullptr,
                                                       tmp1, N, H, H, 1.0f, 1);
  gemm_wmma_f32<<<dim3(ggx, 1), 256, shK128, stream>>>(tmp1, gin_w2, gin_b2, h,
                                                       x1, N, H, H, 1.0f, 0);
  bn_stats<<<bnB, 256, 0, stream>>>(x1, nullptr, nullptr, sum0, sq0, N);

  // -------- sparse attention branch --------
  gemm_wmma_f32<<<dim3(ggx, 1), 256, shK128, stream>>>(h, wq, bq, nullptr, qb, N, H, H, 0.25f, 0);
  gemm_wmma_f32<<<dim3(ggx, 1), 256, shK128, stream>>>(h, wk, bk, nullptr, kb, N, H, H, 1.0f, 0);
  gemm_wmma_f32<<<dim3(ggx, 1), 256, shK128, stream>>>(h, wv, bv, nullptr, vb, N, H, H, 1.0f, 0);
  edge_score<<<eh8, 256, 0, stream>>>(qb, kb, src, dst, sbuf, mbuf, E);
  edge_exp<<<eh8, 256, 0, stream>>>(sbuf, mbuf, src, dbuf, E);
  edge_out<<<eh32, 256, 0, stream>>>(sbuf, dbuf, vb, src, dst, x2, E);
  bn_stats<<<bnB, 256, 0, stream>>>(x2, h, x2, sum1, sq1, N);            // x2 = h + attn

  // -------- combine + FFN + out norm --------
  bn_finalize<<<1, 128, 0, stream>>>(sum0, sq0, nl_g, nl_b, sc0, sh0, N);
  bn_finalize<<<1, 128, 0, stream>>>(sum1, sq1, na_g, na_b, sc1, sh1, N);
  ew_combine<<<ewB, 256, 0, stream>>>(x1, x2, sc0, sh0, sc1, sh1, hcb, (int)NHf);
  gemm_wmma_f32<<<dim3(ggx, 2), 256, shK128, stream>>>(hcb, ffn1_w, ffn1_b, nullptr,
                                                       f1b, N, H, 2 * H, 1.0f, 1);
  gemm_wmma_f32<<<dim3(ggx, 1), 256, shK256, stream>>>(f1b, ffn2_w, ffn2_b, hcb,
                                                       tmp1, N, 2 * H, H, 1.0f, 0);
  bn_stats<<<bnB, 256, 0, stream>>>(tmp1, nullptr, nullptr, sum2, sq2, N);
  bn_finalize<<<1, 128, 0, stream>>>(sum2, sq2, no_g, no_b, sc2, sh2, N);
  ew_apply_bn<<<ewB, 256, 0, stream>>>(tmp1, sc2, sh2, outp, (int)NHf);

  // -------- second tuple output: eattr passthrough --------
  hipMemcpyAsync(outp + NHf, eattr, (size_t)E * H * sizeof(float),
                 hipMemcpyDeviceToDevice, stream);
}